// ParallelMoELayer_7859790152166
// MI455X (gfx1250) — compile-verified
//
#include <hip/hip_runtime.h>
#include <hip/hip_bf16.h>
#include <stdint.h>

// ---------------------------------------------------------------------------
// MoE routing: biased softmax -> top-8 -> renormalize.
// CDNA5 (gfx1250) wave32. One wave per token.
//   - HBM stream: global_load_async_to_lds_b128, triple buffer, depth-2
//     prefetch (ASYNCcnt, in-order completion).
//   - top-8: 8 rounds of DPP row_shr max-reduce + readlane (no DS shuffles).
// ---------------------------------------------------------------------------

#define NE_       128        // experts
#define KSEL      8          // top_k
#define WAVES_PB  8
#define TPB       (WAVES_PB * 32)
#define NBLOCKS   8192
#define NSLOT     3          // LDS buffers per wave (prefetch distance 2)

typedef unsigned int u32;

#define W_ENERGY  0.4f
#define W_LOAD    0.3f
#define W_THERMAL 0.2f
#define W_REROUTE 0.1f

// ---- CDNA5 async global -> LDS copy (16B per lane), tracked by ASYNCcnt ----
__device__ __forceinline__ void async_row16(u32 lds_byte_addr,
                                            const float* __restrict__ gbase,
                                            u32 byte_off) {
  asm volatile("global_load_async_to_lds_b128 %0, %1, %2"
               :
               : "v"(lds_byte_addr), "v"(byte_off), "s"(gbase)
               : "memory");
}
#define WAIT_ASYNC(n) asm volatile("s_wait_asynccnt %0" :: "i"(n) : "memory")

// ---- 32-bit order key: (monotonic float bits & ~0x7F) | (127 - expert) -----
// Top 25 bits order the value; low 7 bits give exact lowest-index tie-break.
// Keys are unique per element. 0 is the identity for max (all live keys > 0).
__device__ __forceinline__ u32 make_key(float x, int low7) {
  u32 u = __float_as_uint(x);
  u = (u & 0x80000000u) ? ~u : (u | 0x80000000u);
  return (u & 0xFFFFFF80u) | (u32)low7;
}

// Wave32 max-reduce via DPP row_shr (rows of 16) + readlane; result uniform.
__device__ __forceinline__ u32 wave_max_u32(u32 v) {
  u32 t;
  t = (u32)__builtin_amdgcn_update_dpp(0, (int)v, 0x111, 0xf, 0xf, true); v = v > t ? v : t;
  t = (u32)__builtin_amdgcn_update_dpp(0, (int)v, 0x112, 0xf, 0xf, true); v = v > t ? v : t;
  t = (u32)__builtin_amdgcn_update_dpp(0, (int)v, 0x114, 0xf, 0xf, true); v = v > t ? v : t;
  t = (u32)__builtin_amdgcn_update_dpp(0, (int)v, 0x118, 0xf, 0xf, true); v = v > t ? v : t;
  u32 a = (u32)__builtin_amdgcn_readlane((int)v, 15);   // max of lanes 0..15
  u32 b = (u32)__builtin_amdgcn_readlane((int)v, 31);   // max of lanes 16..31
  return a > b ? a : b;                                 // scalar
}

// descending compare-exchange
#define CE_DESC(a, b) { u32 hi = (a) > (b) ? (a) : (b); u32 lo = (a) > (b) ? (b) : (a); (a) = hi; (b) = lo; }

__global__ __launch_bounds__(TPB, 2)
void moe_topk_route_kernel(const float* __restrict__ logits,
                           const float* __restrict__ energy_b,
                           const float* __restrict__ load_b,
                           const float* __restrict__ thermal_b,
                           const float* __restrict__ reroute_b,
                           int*   __restrict__ out_idx,
                           float* __restrict__ out_prob,
                           int num_tokens) {
  // Typed float4 shared buffer: direct indexing stays in addrspace(3) so the
  // row read lowers to ds_load_b128 (NOT a flat load through the aperture).
  __shared__ float4 ldsbuf[WAVES_PB][NSLOT][NE_ / 4];

  const int lane = threadIdx.x & 31;
  const int wave = threadIdx.x >> 5;
  const int wgid = blockIdx.x * WAVES_PB + wave;   // global wave id
  const int nwav = gridDim.x * WAVES_PB;

  // Combined bias for this lane's 4 experts (loaded once per wave lifetime).
  const float4 be = ((const float4*)energy_b)[lane];
  const float4 bl = ((const float4*)load_b)[lane];
  const float4 bt = ((const float4*)thermal_b)[lane];
  const float4 br = ((const float4*)reroute_b)[lane];
  float4 bias;
  bias.x = W_ENERGY * be.x + W_LOAD * bl.x + W_THERMAL * bt.x + W_REROUTE * br.x;
  bias.y = W_ENERGY * be.y + W_LOAD * bl.y + W_THERMAL * bt.y + W_REROUTE * br.y;
  bias.z = W_ENERGY * be.z + W_LOAD * bl.z + W_THERMAL * bt.z + W_REROUTE * br.z;
  bias.w = W_ENERGY * be.w + W_LOAD * bl.w + W_THERMAL * bt.w + W_REROUTE * br.w;

  // low-7 key bits for this lane's 4 experts: 127 - (4*lane + slot)
  const int low7base = 127 - 4 * lane;

  // LDS byte address of this lane's 16B chunk in slot 0; slot stride = 512B.
  const u32 lds_lane0 = (u32)(uintptr_t)&ldsbuf[wave][0][lane];

  if (wgid >= num_tokens) return;

  // Prime the pipeline: depth-2 prefetch (token row = 512B, 16B per lane).
  const u32 lane_goff = (u32)lane * 16u;
  async_row16(lds_lane0, logits, (u32)wgid * (NE_ * 4u) + lane_goff);
  if (wgid + nwav < num_tokens)
    async_row16(lds_lane0 + 512u, logits, (u32)(wgid + nwav) * (NE_ * 4u) + lane_goff);

  int slot = 0;
  for (int t = wgid; t < num_tokens; t += nwav) {
    const int t2 = t + 2 * nwav;
    // Completion is in-order on ASYNCcnt: current row is done when at most
    // (#issues after it) remain outstanding.
    if (t2 < num_tokens) {
      int islot = slot + 2; if (islot >= NSLOT) islot -= NSLOT;
      async_row16(lds_lane0 + (u32)islot * 512u, logits,
                  (u32)t2 * (NE_ * 4u) + lane_goff);
      WAIT_ASYNC(2);
    } else if (t + nwav < num_tokens) {
      WAIT_ASYNC(1);
    } else {
      WAIT_ASYNC(0);
    }

    const float4 row = ldsbuf[wave][slot][lane];       // ds_load_b128
    const float x0 = row.x + bias.x;
    const float x1 = row.y + bias.y;
    const float x2 = row.z + bias.z;
    const float x3 = row.w + bias.w;

    u32 a0 = make_key(x0, low7base - 0);
    u32 a1 = make_key(x1, low7base - 1);
    u32 a2 = make_key(x2, low7base - 2);
    u32 a3 = make_key(x3, low7base - 3);

    // sort the lane's 4 keys descending (network: (0,1)(2,3)(0,2)(1,3)(1,2))
    CE_DESC(a0, a1); CE_DESC(a2, a3); CE_DESC(a0, a2); CE_DESC(a1, a3); CE_DESC(a1, a2);

    float wx[KSEL];
    int   we[KSEL];
#pragma unroll
    for (int r = 0; r < KSEL; ++r) {
      const u32 w = wave_max_u32(a0);         // uniform winner key
      const int e = 127 - (int)(w & 127u);    // expert index (scalar)
      // recover the exact adjusted logit from the owning lane
      const int slot2 = e & 3;
      const float s01 = (slot2 & 1) ? x1 : x0;
      const float s23 = (slot2 & 1) ? x3 : x2;
      const float sv  = (slot2 & 2) ? s23 : s01;
      wx[r] = __int_as_float(__builtin_amdgcn_readlane(__float_as_int(sv), e >> 2));
      we[r] = e;
      // winner lane pops its list head (keys unique -> only owner matches)
      const bool won = (a0 == w);
      a0 = won ? a1 : a0;
      a1 = won ? a2 : a1;
      a2 = won ? a3 : a2;
      a3 = won ? 0u : a3;
    }

    // softmax over the 8 selected logits == renormalized top-k probs
    const float m = wx[0];
    float ev[KSEL];
#pragma unroll
    for (int r = 0; r < KSEL; ++r) { ev[r] = __expf(wx[r] - m); }
    const float s01s = (ev[0] + ev[1]) + (ev[2] + ev[3]);
    const float s23s = (ev[4] + ev[5]) + (ev[6] + ev[7]);
    const float inv  = 1.0f / (s01s + s23s);

    // lane r (r < 8) emits result r
    float p_sel = ev[0] * inv;
    int   e_sel = we[0];
#pragma unroll
    for (int r = 1; r < KSEL; ++r) {
      if (lane == r) { p_sel = ev[r] * inv; e_sel = we[r]; }
    }
    if (lane < KSEL) {
      const size_t o = (size_t)t * KSEL + (size_t)lane;
      out_idx[o]  = e_sel;
      out_prob[o] = p_sel;
    }

    slot = (slot + 1 == NSLOT) ? 0 : slot + 1;
  }
}

extern "C" void kernel_launch(void* const* d_in, const int* in_sizes, int n_in,
                              void* d_out, int out_size, void* d_ws, size_t ws_size,
                              hipStream_t stream) {
  (void)n_in; (void)d_ws; (void)ws_size; (void)out_size;
  const float* logits   = (const float*)d_in[0];
  const float* energy_b = (const float*)d_in[1];
  const float* load_b   = (const float*)d_in[2];
  const float* therm_b  = (const float*)d_in[3];
  const float* rerout_b = (const float*)d_in[4];
  // d_in[5] is top_k (== 8, compile-time KSEL)

  const int num_tokens = in_sizes[0] / NE_;

  // Output layout: [T*8] int32 indices, then [T*8] float probs (flat concat).
  int*   out_idx  = (int*)d_out;
  float* out_prob = (float*)d_out + (size_t)num_tokens * KSEL;

  moe_topk_route_kernel<<<dim3(NBLOCKS), dim3(TPB), 0, stream>>>(
      logits, energy_b, load_b, therm_b, rerout_b, out_idx, out_prob, num_tokens);
}